// HyperDecoder_18107582120275
// MI455X (gfx1250) — compile-verified
//
#include <hip/hip_runtime.h>
#include <cstdint>

#define NNODES 100000
#define NEDGES 50000
#define NNZV   1600000
#define DIM    128
#define NLAYER 2

typedef __attribute__((ext_vector_type(2))) float v2f;
typedef __attribute__((ext_vector_type(8))) float v8f;

// ---------------------------------------------------------------- utilities
__global__ void fill_zero_f(float* __restrict__ p, int n) {
    int i = blockIdx.x * blockDim.x + threadIdx.x;
    if (i < n) p[i] = 0.0f;
}

__global__ void degree_count(const long long* __restrict__ nidx,
                             const long long* __restrict__ eidx,
                             float* __restrict__ deg_v,
                             float* __restrict__ deg_e, int nnz) {
    int i = blockIdx.x * blockDim.x + threadIdx.x;
    if (i >= nnz) return;
    atomicAdd(&deg_v[(int)nidx[i]], 1.0f);
    atomicAdd(&deg_e[(int)eidx[i]], 1.0f);
}

__global__ void invert_deg(float* __restrict__ p, int n) {
    int i = blockIdx.x * blockDim.x + threadIdx.x;
    if (i < n) {
        float d = p[i];
        p[i] = (d > 0.0f) ? (1.0f / d) : 0.0f;
    }
}

// ------------------------------------------------- fp32 WMMA GEMM  C = A*W
// A: [M,128] row-major, W: [128,128] row-major, C: [M,128].
// Block = 256 threads = 8 waves; each wave owns one 16x16 tile of C.
// blockIdx.x selects the 16-row stripe; wave id selects the 16-col tile.
// M must be a multiple of 16 (100000/16=6250, 50000/16=3125: exact).
__global__ __launch_bounds__(256) void gemm_f32_wmma(
    const float* __restrict__ A,
    const float* __restrict__ W,
    float* __restrict__ C, int M) {
    const int wave = threadIdx.x >> 5;      // 0..7  -> N tile
    const int lane = threadIdx.x & 31;
    const int l16  = lane & 15;
    const int hi   = lane >> 4;             // 0 for lanes 0-15, 1 for 16-31
    const int row0 = blockIdx.x * 16;
    const int col0 = wave * 16;

    const int aRow = row0 + l16;            // A rows: lanes 0-15 == 16-31
    const int kOff = hi * 2;                // K sub-offset per ISA layout
    const int bCol = col0 + l16;

    v8f acc = {};
    const float* aBase = A + (size_t)aRow * DIM;
#pragma unroll 4
    for (int k = 0; k < DIM; k += 4) {
        // A 16x4 frag: VGPR0=K(k+kOff), VGPR1=K(k+kOff+1)
        v2f a = *(const v2f*)(aBase + k + kOff);
        // B 4x16 frag (mirrored layout): VGPR0=W[k+kOff][col], VGPR1=W[k+kOff+1][col]
        v2f b;
        b.x = W[(size_t)(k + kOff) * DIM + bCol];
        b.y = W[(size_t)(k + kOff + 1) * DIM + bCol];
        acc = __builtin_amdgcn_wmma_f32_16x16x4_f32(
            /*neg_a=*/false, a, /*neg_b=*/false, b,
            /*c_mod=*/(short)0, acc, /*reuse_a=*/false, /*reuse_b=*/false);
    }

    // C/D layout: VGPR v holds row (hi*8 + v), col = l16
    const int rBase = row0 + hi * 8;
    const int cCol  = col0 + l16;
#pragma unroll
    for (int v = 0; v < 8; ++v)
        C[(size_t)(rBase + v) * DIM + cCol] = acc[v];
}

// ------------------------------------------- scatter-add of 128-float rows
// One wave (32 lanes) per incidence; each lane handles a float4 slice.
__global__ __launch_bounds__(256) void scatter_add_rows(
    const float* __restrict__ src,          // [S, 128]
    const long long* __restrict__ gather_idx,
    const long long* __restrict__ scatter_idx,
    float* __restrict__ dst,                // [T, 128]
    int nnz) {
    int t = blockIdx.x * blockDim.x + threadIdx.x;
    int inc  = t >> 5;
    int lane = t & 31;
    if (inc >= nnz) return;
    int s = (int)gather_idx[inc];
    int d = (int)scatter_idx[inc];
    const float4 v = *(const float4*)(src + (size_t)s * DIM + lane * 4);
    float* p = dst + (size_t)d * DIM + lane * 4;
    atomicAdd(p + 0, v.x);
    atomicAdd(p + 1, v.y);
    atomicAdd(p + 2, v.z);
    atomicAdd(p + 3, v.w);
}

// h = m * inv_e[row] + e1 + b[col]   (in-place into m)
__global__ void combine_h(float* __restrict__ m,
                          const float* __restrict__ e1,
                          const float* __restrict__ inv_e,
                          const float* __restrict__ bias, int n) {
    int i = blockIdx.x * blockDim.x + threadIdx.x;
    if (i >= n) return;
    int row = i >> 7;
    int col = i & (DIM - 1);
    m[i] = m[i] * inv_e[row] + e1[i] + bias[col];
}

// out = PReLU(acc * inv_v[row]) with shared slope *a_ptr
__global__ void prelu_scale(const float* __restrict__ acc,
                            const float* __restrict__ inv_v,
                            const float* __restrict__ a_ptr,
                            float* __restrict__ out, int n) {
    int i = blockIdx.x * blockDim.x + threadIdx.x;
    if (i >= n) return;
    float alpha = *a_ptr;
    float v = acc[i] * inv_v[i >> 7];
    out[i] = (v >= 0.0f) ? v : alpha * v;
}

// ---------------------------------------------------------------- launcher
extern "C" void kernel_launch(void* const* d_in, const int* in_sizes, int n_in,
                              void* d_out, int out_size, void* d_ws, size_t ws_size,
                              hipStream_t stream) {
    const float* x  = (const float*)d_in[0];           // [N, 128]
    const float* e  = (const float*)d_in[1];           // [E, 128]
    const long long* hei = (const long long*)d_in[2];  // [2, NNZ] int64
    const float* Wn = (const float*)d_in[5];           // [L, 128, 128]
    const float* We = (const float*)d_in[6];           // [L, 128, 128]
    const float* bb = (const float*)d_in[7];           // [L, 128]
    const float* aa = (const float*)d_in[8];           // scalar slope
    float* out = (float*)d_out;

    const long long* nidx = hei;
    const long long* eidx = hei + NNZV;

    // workspace layout (floats)
    float* ws    = (float*)d_ws;
    float* inv_e = ws;                                  // NEDGES
    float* inv_v = inv_e + NEDGES;                      // NNODES (contiguous w/ inv_e)
    float* xcur  = inv_v + NNODES;                      // N*D
    float* x1    = xcur + (size_t)NNODES * DIM;         // N*D (GEMM out / node accum)
    float* m     = x1   + (size_t)NNODES * DIM;         // E*D (edge accum / h)
    float* e1    = m    + (size_t)NEDGES * DIM;         // E*D (e @ We)

    const int TB = 256;
    const int nDeg = NEDGES + NNODES;

    // degrees -> inverse degrees
    fill_zero_f<<<(nDeg + TB - 1) / TB, TB, 0, stream>>>(inv_e, nDeg);
    degree_count<<<(NNZV + TB - 1) / TB, TB, 0, stream>>>(nidx, eidx, inv_v, inv_e, NNZV);
    invert_deg<<<(nDeg + TB - 1) / TB, TB, 0, stream>>>(inv_e, nDeg);

    const int scatterBlocks = (NNZV * 32) / TB;         // 51.2M threads exact
    const int nND = NNODES * DIM;                       // 12.8M, /256 exact
    const int nED = NEDGES * DIM;                       // 6.4M, /256 exact

    for (int l = 0; l < NLAYER; ++l) {
        const float* xin = (l == 0) ? x : xcur;
        const float* Wl  = Wn + (size_t)l * DIM * DIM;
        const float* Vl  = We + (size_t)l * DIM * DIM;
        const float* bl  = bb + (size_t)l * DIM;

        // x1 = x @ Wn[l]
        gemm_f32_wmma<<<NNODES / 16, TB, 0, stream>>>(xin, Wl, x1, NNODES);
        // m = segment_sum(x1[nidx] -> eidx)
        fill_zero_f<<<nED / TB, TB, 0, stream>>>(m, nED);
        scatter_add_rows<<<scatterBlocks, TB, 0, stream>>>(x1, nidx, eidx, m, NNZV);
        // e1 = e @ We[l]
        gemm_f32_wmma<<<NEDGES / 16, TB, 0, stream>>>(e, Vl, e1, NEDGES);
        // h = m * inv_e + e1 + b[l]   (into m)
        combine_h<<<nED / TB, TB, 0, stream>>>(m, e1, inv_e, bl, nED);
        // x1 = segment_sum(h[eidx] -> nidx)
        fill_zero_f<<<nND / TB, TB, 0, stream>>>(x1, nND);
        scatter_add_rows<<<scatterBlocks, TB, 0, stream>>>(m, eidx, nidx, x1, NNZV);
        // x = PReLU(x1 * inv_v)
        float* xo = (l == NLAYER - 1) ? out : xcur;
        prelu_scale<<<nND / TB, TB, 0, stream>>>(x1, inv_v, aa, xo, nND);
    }
}